// SourceModulePCPH_50672024158740
// MI455X (gfx1250) — compile-verified
//
#include <hip/hip_runtime.h>
#include <math.h>

#define SR_F     48000.0f
#define T_LEN    96000
#define B_N      16
#define NFRAMES  200
#define HOP      480
#define PI_F     3.14159265358979323846f
#define TWO_PI_D 6.283185307179586476925287

typedef __attribute__((ext_vector_type(2))) float v2f;
typedef __attribute__((ext_vector_type(8))) float v8f;

// ---------------------------------------------------------------- constants
__global__ void k_ir(float* __restrict__ ir) {
    __shared__ float red[128];
    int i = threadIdx.x;
    float v = 0.f;
    if (i < 127) {
        float t = -0.5f + (float)i * (1.0f / 126.0f);
        float chirp = sinf(50.0f * t * t);
        float win = 0.5f * (1.0f - cosf(2.0f * PI_F * (float)i / 127.0f));
        v = chirp * win;
    }
    red[i] = v * v;
    __syncthreads();
    for (int s = 64; s > 0; s >>= 1) {
        if (i < s) red[i] += red[i + s];
        __syncthreads();
    }
    float norm = sqrtf(red[0]) + 1e-8f;
    ir[i] = (i < 127) ? (v / norm) : 0.f;
}

// ---------------------------------------------------------------- PCHIP slopes
__device__ __forceinline__ float signf(float x) {
    return (x > 0.f) ? 1.f : ((x < 0.f) ? -1.f : 0.f);
}
__device__ __forceinline__ float endslope(float d0, float d1) {
    float d = 0.5f * (3.f * d0 - d1);
    if (signf(d) != signf(d0)) d = 0.f;
    else if ((signf(d0) != signf(d1)) && (fabsf(d) > 3.f * fabsf(d0))) d = 3.f * d0;
    return d;
}

__global__ void k_pchip(const float* __restrict__ f0, float* __restrict__ dv) {
    int b = blockIdx.x, f = threadIdx.x;
    if (f >= NFRAMES) return;
    const float* y = f0 + b * NFRAMES;
    float d;
    if (f == 0) {
        d = endslope(y[1] - y[0], y[2] - y[1]);
    } else if (f == NFRAMES - 1) {
        d = endslope(y[NFRAMES - 1] - y[NFRAMES - 2], y[NFRAMES - 2] - y[NFRAMES - 3]);
    } else {
        float dl = y[f] - y[f - 1], dr = y[f + 1] - y[f];
        float prod = dl * dr;
        d = (prod > 0.f) ? (2.0f * prod / (dl + dr + 1e-30f)) : 0.f;
    }
    dv[b * NFRAMES + f] = d;
}

__device__ __forceinline__ float pchip_eval(const float* y, const float* dv, int t) {
    int i = t / HOP;
    if (i > NFRAMES - 2) i = NFRAMES - 2;
    float u = (float)((double)t / (double)HOP - (double)i);
    float u2 = u * u, u3 = u2 * u;
    float h00 = 2.f * u3 - 3.f * u2 + 1.f;
    float h10 = u3 - 2.f * u2 + u;
    float h01 = -2.f * u3 + 3.f * u2;
    float h11 = u3 - u2;
    return h00 * y[i] + h10 * dv[i] + h01 * y[i + 1] + h11 * dv[i + 1];
}

// ---------------------------------------------------------------- phase scan (f64 cumsum per batch)
__global__ void k_phase(const float* __restrict__ f0, const float* __restrict__ init_phase,
                        const float* __restrict__ dv,
                        float* __restrict__ f0u_out, float* __restrict__ phase_out) {
    int b = blockIdx.x, tid = threadIdx.x;
    __shared__ double sd[256];
    __shared__ double carry;
    if (tid == 0) carry = 0.0;
    __syncthreads();
    const float* y = f0 + b * NFRAMES;
    const float* dvb = dv + b * NFRAMES;
    for (int chunk = 0; chunk < T_LEN / 256; ++chunk) {
        int t = chunk * 256 + tid;
        float f0v = pchip_eval(y, dvb, t);
        float incf = f0v / SR_F;
        if (t == 0) incf += init_phase[b];
        sd[tid] = (double)incf;
        __syncthreads();
        for (int s = 1; s < 256; s <<= 1) {
            double p = (tid >= s) ? sd[tid - s] : 0.0;
            __syncthreads();
            sd[tid] += p;
            __syncthreads();
        }
        double tot = carry + sd[tid];
        double ph = fmod(tot * TWO_PI_D, TWO_PI_D);
        f0u_out[b * T_LEN + t] = f0v;
        phase_out[b * T_LEN + t] = (float)ph;
        __syncthreads();
        if (tid == 0) carry += sd[255];
        __syncthreads();
    }
}

// ---------------------------------------------------------------- harmonic excitation + noise select
__global__ void k_excite(const float* __restrict__ f0u, const float* __restrict__ phase,
                         const float* __restrict__ n1, const float* __restrict__ n2,
                         float* __restrict__ sig0, float* __restrict__ fnoise) {
    int idx = blockIdx.x * blockDim.x + threadIdx.x;
    if (idx >= B_N * T_LEN) return;
    float f0v = f0u[idx];
    float ph = phase[idx];
    float voiced = (f0v > 1.0f) ? 1.f : 0.f;
    float safe = fmaxf(f0v, 1e-5f);
    float N = floorf(24000.0f / safe);
    float half = 0.5f * ph;
    float den = sinf(half);
    if (fabsf(den) < 1e-6f) den = (den >= 0.f) ? 1e-6f : -1e-6f;
    float harm = sinf(N * half) * sinf((N + 1.0f) * half) / den;
    float amp = 0.1f * sqrtf(2.0f / fmaxf(N, 1.0f));
    sig0[idx] = harm * amp * voiced;
    float decay = expf(-ph / (2.0f * PI_F * 0.87f));
    fnoise[idx] = (voiced > 0.5f) ? n1[idx] * decay * 0.003f : n2[idx] * (0.1f / 3.0f);
}

// ---------------------------------------------------------------- dispersion FIR as Toeplitz GEMM (WMMA f32 16x16x4)
// Out[t0+j, b] = sum_k ir_ext[k - j] * x[b, t0 - 63 + k],  k in [0,144) (padded)
#define STG_COLS   256
#define STG_STRIDE 257
__global__ void k_disperse(const float* __restrict__ sig0, const float* __restrict__ ir,
                           float* __restrict__ out) {
    __shared__ float stage[16 * STG_STRIDE];
    __shared__ float irext[160];
    int tid = threadIdx.x;
    int t0blk = blockIdx.x * 128;

    if (tid < 160) {
        int k = tid - 15;
        irext[tid] = (k >= 0 && k < 127) ? ir[k] : 0.f;
    }
    // coalesced stage of x[b, t0blk-63 .. t0blk+192]
    for (int it = 0; it < 16; ++it) {
        int flat = it * 256 + tid;
        int row = flat >> 8;    // batch
        int col = flat & 255;
        int g = t0blk - 63 + col;
        float v = (g >= 0 && g < T_LEN) ? sig0[row * T_LEN + g] : 0.f;
        stage[row * STG_STRIDE + col] = v;
    }
    __syncthreads();

    int wv = tid >> 5;          // wave id (tile id), wave32
    int lane = tid & 31;
    int m = lane & 15;          // A row (time offset j) / B col (batch)
    int hi = lane >> 4;
    int qsel = hi * 2;          // K placement: lanes 0-15 hold K=0,1; 16-31 hold K=2,3

    v8f acc = {0.f, 0.f, 0.f, 0.f, 0.f, 0.f, 0.f, 0.f};
    const float* brow = &stage[m * STG_STRIDE + wv * 16];
    #pragma unroll 4
    for (int c = 0; c < 36; ++c) {
        int ai = 15 + 4 * c + qsel - m;     // Toeplitz A: irext[15 + k - j]
        v2f a; a.x = irext[ai]; a.y = irext[ai + 1];
        int bc = 4 * c + qsel;
        v2f bb; bb.x = brow[bc]; bb.y = brow[bc + 1];
        acc = __builtin_amdgcn_wmma_f32_16x16x4_f32(
            false, a, false, bb, (short)0, acc, false, false);
    }
    int t0 = t0blk + wv * 16;
    int j0 = hi * 8;
    float* obase = out + m * T_LEN + t0;    // D: N=lane&15 -> batch, M=r(+8*hi) -> time
    #pragma unroll
    for (int r = 0; r < 8; ++r) obase[j0 + r] = acc[r];
}

// ---------------------------------------------------------------- tilt MLP -> alpha
__global__ void k_alpha(const float* __restrict__ f0u,
                        const float* __restrict__ w1, const float* __restrict__ b1,
                        const float* __restrict__ w2, const float* __restrict__ b2,
                        float* __restrict__ alpha) {
    __shared__ float sw1[64], sb1[64], sw2[64];
    int tid = threadIdx.x;
    if (tid < 64) { sw1[tid] = w1[tid]; sb1[tid] = b1[tid]; sw2[tid] = w2[tid]; }
    __syncthreads();
    int idx = blockIdx.x * 256 + tid;
    if (idx >= B_N * T_LEN) return;
    float f0v = f0u[idx];
    float x = log2f(f0v * (10.0f / 24000.0f) + 1.0f);
    float accv = b2[0];
    #pragma unroll 8
    for (int o = 0; o < 64; ++o) {
        float h = fmaf(sw1[o], x, sb1[o]);
        h = (h >= 0.f) ? h : 0.2f * h;
        accv = fmaf(sw2[o], h, accv);
    }
    float sg = 1.0f / (1.0f + expf(-accv));
    float voiced = (f0v > 1.0f) ? 1.f : 0.f;
    alpha[idx] = sg * 0.98f * voiced;
}

// ---------------------------------------------------------------- first-order IIR via pair scan, in-place
__global__ void k_iir(const float* __restrict__ alpha, float* __restrict__ sig,
                      float* __restrict__ state_out) {
    int b = blockIdx.x, tid = threadIdx.x;
    __shared__ float sa[256], sv[256];
    __shared__ float carry;
    if (tid == 0) carry = 0.f;
    __syncthreads();
    for (int chunk = 0; chunk < T_LEN / 256; ++chunk) {
        int t = chunk * 256 + tid;
        float a = alpha[b * T_LEN + t];
        float x = (1.0f - a) * sig[b * T_LEN + t];
        sa[tid] = a; sv[tid] = x;
        __syncthreads();
        for (int s = 1; s < 256; s <<= 1) {
            float pa = 0.f, pb = 0.f;
            if (tid >= s) { pa = sa[tid - s]; pb = sv[tid - s]; }
            __syncthreads();
            if (tid >= s) { sv[tid] = fmaf(sa[tid], pb, sv[tid]); sa[tid] *= pa; }
            __syncthreads();
        }
        float yv = fmaf(sa[tid], carry, sv[tid]);
        sig[b * T_LEN + t] = yv;
        __syncthreads();
        if (tid == 255) carry = yv;
        __syncthreads();
    }
    if (tid == 0) state_out[b] = carry;
}

// ---------------------------------------------------------------- time-varying sinc FIR pair + tanh
__global__ void k_final(const float* __restrict__ y, const float* __restrict__ fnoise,
                        const float* __restrict__ f0u,
                        const float* __restrict__ cut_w, const float* __restrict__ cut_b,
                        const float* __restrict__ lin_w, float* __restrict__ exc) {
    int idx = blockIdx.x * blockDim.x + threadIdx.x;
    if (idx >= B_N * T_LEN) return;
    int b = idx / T_LEN;
    int t = idx - b * T_LEN;
    float f0v = f0u[idx];
    float c = 1.0f / (1.0f + expf(-(cut_w[0] * (f0v / 24000.0f) + cut_b[0])));
    float lp[31];
    float lpsum = 0.f;
    #pragma unroll
    for (int i = 0; i < 31; ++i) {
        float n = (float)(i - 15);
        float win = 0.54f + 0.46f * cosf(2.0f * PI_F * n / 31.0f);
        float xa = c * n;
        float s = (i == 15) ? 1.0f : sinf(PI_F * xa) / (PI_F * xa);
        float v = c * s * win;
        lp[i] = v; lpsum += v;
    }
    float inv_lp = 1.0f / (lpsum + 1e-8f);
    float hpfs = 0.f;
    #pragma unroll
    for (int i = 0; i < 31; ++i) {
        lp[i] *= inv_lp;
        float hp = ((i == 15) ? 1.0f : 0.0f) - lp[i];
        float flip = ((i & 1) == 1) ? 1.0f : -1.0f;   // n = i-15 even <=> i odd
        hpfs += hp * flip;
    }
    float inv_hp = 1.0f / (hpfs + 1e-8f);
    float acc = 0.f;
    const float* yb = y + b * T_LEN;
    const float* nb = fnoise + b * T_LEN;
    #pragma unroll
    for (int i = 0; i < 31; ++i) {
        int ti = t + i - 30;
        float svv = 0.f, nvv = 0.f;
        if (ti >= 0) { svv = yb[ti]; nvv = nb[ti]; }
        float hp = (((i == 15) ? 1.0f : 0.0f) - lp[i]) * inv_hp;
        acc += svv * lp[i] + nvv * hp;
    }
    exc[idx] = tanhf(lin_w[0] * acc);
}

// ---------------------------------------------------------------- launch
extern "C" void kernel_launch(void* const* d_in, const int* in_sizes, int n_in,
                              void* d_out, int out_size, void* d_ws, size_t ws_size,
                              hipStream_t stream) {
    const float* f0         = (const float*)d_in[0];
    const float* init_phase = (const float*)d_in[1];
    const float* noise1     = (const float*)d_in[2];
    const float* noise2     = (const float*)d_in[3];
    const float* w1         = (const float*)d_in[4];
    const float* b1         = (const float*)d_in[5];
    const float* w2         = (const float*)d_in[6];
    const float* b2         = (const float*)d_in[7];
    const float* cut_w      = (const float*)d_in[8];
    const float* cut_b      = (const float*)d_in[9];
    const float* lin_w      = (const float*)d_in[10];

    float* ws = (float*)d_ws;
    const size_t BT = (size_t)B_N * T_LEN;
    float* irb    = ws;                // 128
    float* dvb    = ws + 128;          // B*200
    float* f0u    = ws + 3328;         // BT
    float* bufA   = f0u + BT;          // phase -> sigd -> y (aliased, lifetimes disjoint)
    float* bufB   = bufA + BT;         // sig0 -> alpha
    float* fnoise = bufB + BT;         // BT

    float* exc   = (float*)d_out;
    float* state = exc + BT;

    const int nel = (int)BT;           // 1,536,000
    k_ir<<<1, 128, 0, stream>>>(irb);
    k_pchip<<<B_N, 256, 0, stream>>>(f0, dvb);
    k_phase<<<B_N, 256, 0, stream>>>(f0, init_phase, dvb, f0u, bufA);
    k_excite<<<nel / 256, 256, 0, stream>>>(f0u, bufA, noise1, noise2, bufB, fnoise);
    k_disperse<<<T_LEN / 128, 256, 0, stream>>>(bufB, irb, bufA);
    k_alpha<<<nel / 256, 256, 0, stream>>>(f0u, w1, b1, w2, b2, bufB);
    k_iir<<<B_N, 256, 0, stream>>>(bufB, bufA, state);
    k_final<<<nel / 256, 256, 0, stream>>>(bufA, fnoise, f0u, cut_w, cut_b, lin_w, exc);
}